// PreHTSK_49976239456756
// MI455X (gfx1250) — compile-verified
//
#include <hip/hip_runtime.h>
#include <hip/hip_bf16.h>
#include <cstddef>
#include <cstdint>

typedef __attribute__((ext_vector_type(2))) float v2f;
typedef __attribute__((ext_vector_type(8))) float v8f;
typedef unsigned int u32x4 __attribute__((ext_vector_type(4)));
typedef int i32x4 __attribute__((ext_vector_type(4)));
typedef int i32x8 __attribute__((ext_vector_type(8)));

#define B_      2048
#define IN_DIM  256
#define NR      128
#define OUT_DIM 64
#define DD      ((IN_DIM + 1) * NR)   // 32896
#define TAILOFF (IN_DIM * NR)         // 32768
#define HCONST  0.5f
#define EPS_    1e-8f
#define LN_EPS_ 1e-5f

#define RG   8            // rule-groups (grid.y split of K for occupancy)
#define RPG  (NR / RG)    // 16 rules per group
#define BT   32           // batch rows per workgroup in the WMMA GEMM
#define ROWS 8            // batch rows per block in the firing kernel

#define SW_STRIDE 132     // 128 data floats + 4 pad floats per o-row in LDS
#define TILE_FLOATS (OUT_DIM * SW_STRIDE)
#define GEMM_LDS_BYTES (2 * TILE_FLOATS * 4)   // double-buffered: 67584 B
#define NSTAGE (2 * RPG)                       // 2 chunks x 16 rules

#if __has_builtin(__builtin_amdgcn_tensor_load_to_lds)
#define HAVE_TDM 1
#else
#define HAVE_TDM 0
#endif

#if __has_builtin(__builtin_amdgcn_s_wait_tensorcnt)
#define WAIT_TENSOR0() __builtin_amdgcn_s_wait_tensorcnt(0)
#else
#define WAIT_TENSOR0() asm volatile("s_wait_tensorcnt 0x0" ::: "memory")
#endif

// ---------------- wave32 / block reductions ----------------
__device__ __forceinline__ float waveReduceSum(float v) {
#pragma unroll
  for (int off = 16; off > 0; off >>= 1) v += __shfl_xor(v, off, 32);
  return v;
}
__device__ __forceinline__ float waveReduceMax(float v) {
#pragma unroll
  for (int off = 16; off > 0; off >>= 1) v = fmaxf(v, __shfl_xor(v, off, 32));
  return v;
}
__device__ __forceinline__ float blockSum128(float v, float* red, int tid) {
  v = waveReduceSum(v);
  if ((tid & 31) == 0) red[tid >> 5] = v;
  __syncthreads();
  float s = red[0] + red[1] + red[2] + red[3];
  __syncthreads();
  return s;
}
__device__ __forceinline__ float blockMax128(float v, float* red, int tid) {
  v = waveReduceMax(v);
  if ((tid & 31) == 0) red[tid >> 5] = v;
  __syncthreads();
  float s = fmaxf(fmaxf(red[0], red[1]), fmaxf(red[2], red[3]));
  __syncthreads();
  return s;
}

// ---------------- kernel 1a: scale = H/sigma^2 + eps ----------------
__global__ void k_scale(const float* __restrict__ sigmas, float* __restrict__ scaleArr) {
  int idx = blockIdx.x * 256 + threadIdx.x;
  if (idx < IN_DIM * NR) {
    float s = sigmas[idx];
    scaleArr[idx] = HCONST / (s * s) + EPS_;
  }
}

// ---------------- kernel 1b: G = gamma (elementwise) * W  [64][32896] ----------------
__global__ void k_gw(const float* __restrict__ lnGamma, const float* __restrict__ W,
                     float* __restrict__ G) {
  const int d = blockIdx.x * 256 + threadIdx.x;
  const int o = blockIdx.y;
  if (d < DD) {
    size_t idx = (size_t)o * DD + d;
    G[idx] = lnGamma[d] * W[idx];
  }
}

// ---------------- kernel 2: A2[o] = sum gamma*W, C[o] = sum beta*W + b ----------------
__global__ void k_wsum(const float* __restrict__ lnGamma, const float* __restrict__ lnBeta,
                       const float* __restrict__ W, const float* __restrict__ bvec,
                       float* __restrict__ A2, float* __restrict__ Cc) {
  __shared__ float redG[8], redB[8];
  const int o = blockIdx.x;
  const int tid = threadIdx.x;
  const float* Wr = W + (size_t)o * DD;
  float ga = 0.f, cb = 0.f;
  for (int d = tid; d < DD; d += 256) {
    float w = Wr[d];
    ga = fmaf(lnGamma[d], w, ga);
    cb = fmaf(lnBeta[d], w, cb);
  }
  ga = waveReduceSum(ga);
  cb = waveReduceSum(cb);
  if ((tid & 31) == 0) { redG[tid >> 5] = ga; redB[tid >> 5] = cb; }
  __syncthreads();
  if (tid == 0) {
    float sg = 0.f, sb = 0.f;
#pragma unroll
    for (int i = 0; i < 8; ++i) { sg += redG[i]; sb += redB[i]; }
    A2[o] = sg;
    Cc[o] = sb + bvec[o];
  }
}

// ---------------- kernel 3: firing levels + softmax + closed-form LN stats ----------------
__global__ void k_firing(const float* __restrict__ X, const float* __restrict__ centers,
                         const float* __restrict__ scaleArr,
                         float* __restrict__ frsOut, float* __restrict__ muOut,
                         float* __restrict__ istdOut) {
  __shared__ float Xs[ROWS][IN_DIM];
  __shared__ float red[4];
  const int tid = threadIdx.x;         // 0..127, rule index
  const int r = tid;
  const int rowBase = blockIdx.x * ROWS;
#pragma unroll
  for (int j = 0; j < (ROWS * IN_DIM) / 128; ++j) {
    int idx = tid + j * 128;
    Xs[idx >> 8][idx & 255] = X[(size_t)(rowBase + (idx >> 8)) * IN_DIM + (idx & 255)];
  }
  __syncthreads();

  float acc[ROWS];
#pragma unroll
  for (int j = 0; j < ROWS; ++j) acc[j] = 0.f;
  for (int i = 0; i < IN_DIM; ++i) {
    float c = centers[i * NR + r];
    float s = scaleArr[i * NR + r];
#pragma unroll
    for (int j = 0; j < ROWS; ++j) {
      float d = Xs[j][i] - c;
      acc[j] = fmaf(d * s, d, acc[j]);
    }
  }
  for (int j = 0; j < ROWS; ++j) {
    float fr = -acc[j] * (1.0f / IN_DIM);
    float mx = blockMax128(fr, red, tid);
    float e  = expf(fr - mx);
    float se = blockSum128(e, red, tid);
    float frs = e / se;
    float se2 = blockSum128(e * e, red, tid);
    float frs2sum = se2 / (se * se);          // sum of frs^2
    float x0 = Xs[j][r], x1 = Xs[j][r + 128];
    float sx  = blockSum128(x0 + x1, red, tid);
    float sx2 = blockSum128(x0 * x0 + x1 * x1, red, tid);
    frsOut[(size_t)(rowBase + j) * NR + r] = frs;
    if (tid == 0) {
      // softmax sums to 1:  sum feats = sx + 1 ; sum feats^2 = (sx2+1)*frs2sum
      float mu = (sx + 1.0f) / (float)DD;
      float ms = (sx2 + 1.0f) * frs2sum / (float)DD;
      float var = ms - mu * mu;
      muOut[rowBase + j] = mu;
      istdOut[rowBase + j] = rsqrtf(var + LN_EPS_);
    }
  }
}

#if HAVE_TDM
// Issue one TDM 2D tile load: 64 rows (o) x 128 floats, row stride DD elements,
// LDS destination padded 4 DWORDs per 128 DWORDs -> [o][SW_STRIDE] layout.
__device__ __forceinline__ void tdm_issue_tile(const float* gsrc, float* ldsDst) {
  unsigned ldsAddr = (unsigned)(uintptr_t)(__attribute__((address_space(3))) float*)ldsDst;
  unsigned long long ga = (unsigned long long)(uintptr_t)gsrc;
  u32x4 g0;
  g0[0] = 1u;                                                // count=1 (valid), user mode
  g0[1] = ldsAddr;                                           // lds_addr (bytes)
  g0[2] = (unsigned)(ga & 0xFFFFFFFFu);                      // global_addr[31:0]
  g0[3] = (unsigned)((ga >> 32) & 0x01FFFFFFu) | (2u << 30); // global_addr[56:32] | type=2
  i32x8 g1;
  g1[0] = (2 << 16) | (1 << 20) | (6 << 22) | (3 << 25);     // data=4B, pad_en, interval=128dw, pad=4dw
  g1[1] = (int)((DD & 0xFFFF) << 16);                        // tensor_dim0[15:0]
  g1[2] = (int)((DD >> 16) | (OUT_DIM << 16));               // tensor_dim0[31:16] | tensor_dim1=64
  g1[3] = (int)(128 << 16);                                  // tile_dim0 = 128
  g1[4] = OUT_DIM;                                           // tile_dim1 = 64, tile_dim2 = 0
  g1[5] = (int)DD;                                           // tensor_dim0_stride[31:0]
  g1[6] = 0;
  g1[7] = 0;
  i32x4 gz = {0, 0, 0, 0};
#if __clang_major__ >= 23
  i32x8 gz8 = {0, 0, 0, 0, 0, 0, 0, 0};
  __builtin_amdgcn_tensor_load_to_lds(g0, g1, gz, gz, gz8, 0);
#else
  __builtin_amdgcn_tensor_load_to_lds(g0, g1, gz, gz, 0);
#endif
}
#endif

// ---------------- kernel 4: WMMA GEMM, double-buffered TDM weight tiles ----------------
// partial[g][b][o] = sum_{r in group g} frs[b,r] * (X[b,:] . G[o, r*256 : r*256+256])
__global__ void k_gemm_wmma(const float* __restrict__ X, const float* __restrict__ frsArr,
                            const float* __restrict__ G, float* __restrict__ partial) {
  extern __shared__ float sW[];               // 2 x TILE_FLOATS (67.6 KB)
  const int tid  = threadIdx.x;               // 256 threads = 8 waves
  const int lane = tid & 31;
  const int wave = tid >> 5;
  const int n    = lane & 15;                 // N index (and M index of A frag)
  const int h    = lane >> 4;                 // lane half
  const int wrow = (wave & 1) * 16;
  const int otile = (wave >> 1) * 16;
  const int b0   = blockIdx.x * BT;
  const int rbase = blockIdx.y * RPG;
  const int row  = b0 + wrow + n;
  const float* Xr = X + (size_t)row * IN_DIM;
  const int oc = otile + n;

  v8f acc = {};
  float2 xreg[32];

#if HAVE_TDM
  if (wave == 0) tdm_issue_tile(G + rbase * IN_DIM, sW);   // prologue: stage 0 -> buf 0
#endif

  for (int s = 0; s < NSTAGE; ++s) {
    const int chunk = s >> 4;                 // RPG == 16
    const int rr = s & 15;
    if (rr == 0) {
      // hoist this lane's A-fragment X data for the 128-wide chunk (reused by all rules)
      const float* Xc = Xr + chunk * 128 + 2 * h;
#pragma unroll
      for (int t = 0; t < 32; ++t) xreg[t] = *(const float2*)(Xc + 4 * t);
    }
    const float frv = frsArr[(size_t)row * NR + rbase + rr];
    float* buf = sW + (s & 1) * TILE_FLOATS;
#if HAVE_TDM
    if (wave == 0) WAIT_TENSOR0();            // load for stage s has landed
    __syncthreads();                          // publish buf[s&1]; back buffer now free
    if (wave == 0 && s + 1 < NSTAGE) {        // prefetch next tile, overlapped with WMMA
      const int s1 = s + 1;
      const int kb1 = (rbase + (s1 & 15)) * IN_DIM + (s1 >> 4) * 128;
      tdm_issue_tile(G + kb1, sW + (s1 & 1) * TILE_FLOATS);
    }
#else
    const int kbase = (rbase + rr) * IN_DIM + chunk * 128;
    __syncthreads();                          // previous tile fully consumed
#pragma unroll
    for (int j = 0; j < 8; ++j) {             // cooperative fallback copy
      int v = tid + j * 256;
      int o = v >> 5;
      int i4 = (v & 31) << 2;
      *(float4*)&buf[o * SW_STRIDE + i4] = *(const float4*)(G + (size_t)o * DD + kbase + i4);
    }
    __syncthreads();                          // publish tile
#endif
    const float* sWrow = buf + oc * SW_STRIDE + 2 * h;
#pragma unroll
    for (int t = 0; t < 32; ++t) {
      v2f a; a.x = frv * xreg[t].x; a.y = frv * xreg[t].y;   // A[m][2h..2h+1]
      float2 bv = *(const float2*)(sWrow + 4 * t);           // B[2h..2h+1][n]
      v2f bb; bb.x = bv.x; bb.y = bv.y;
      acc = __builtin_amdgcn_wmma_f32_16x16x4_f32(
          /*neg_a=*/false, a, /*neg_b=*/false, bb,
          /*c_mod=*/(short)0, acc, /*reuse_a=*/false, /*reuse_b=*/false);
    }
  }
  // D layout: VGPR j holds M = j + 8*h, N = n
  float* P = partial + (size_t)blockIdx.y * (B_ * OUT_DIM);
#pragma unroll
  for (int j = 0; j < 8; ++j) {
    int rj = b0 + wrow + j + 8 * h;
    P[(size_t)rj * OUT_DIM + oc] = acc[j];
  }
}

// ---------------- kernel 5: combine partials + tail + LN-folded affine ----------------
__global__ void k_finalize(const float* __restrict__ partial, const float* __restrict__ frsArr,
                           const float* __restrict__ G, const float* __restrict__ muArr,
                           const float* __restrict__ istdArr, const float* __restrict__ A2,
                           const float* __restrict__ Cc, float* __restrict__ Out) {
  const int idx = blockIdx.x * 256 + threadIdx.x;   // < B*OUT_DIM
  const int b = idx >> 6;
  const int o = idx & 63;
  float s = 0.f;
#pragma unroll
  for (int g = 0; g < RG; ++g) s += partial[(size_t)g * (B_ * OUT_DIM) + idx];
  const float* fr = frsArr + (size_t)b * NR;
  const float* gw = G + (size_t)o * DD + TAILOFF;   // tail weights gamma*W
  float t = 0.f;
#pragma unroll 8
  for (int r = 0; r < NR; ++r) t = fmaf(fr[r], gw[r], t);
  Out[idx] = istdArr[b] * (s + t - muArr[b] * A2[o]) + Cc[o];
}

// ---------------- host launch ----------------
extern "C" void kernel_launch(void* const* d_in, const int* in_sizes, int n_in,
                              void* d_out, int out_size, void* d_ws, size_t ws_size,
                              hipStream_t stream) {
  const float* X       = (const float*)d_in[0];
  const float* centers = (const float*)d_in[1];
  const float* sigmas  = (const float*)d_in[2];
  const float* lnGamma = (const float*)d_in[3];
  const float* lnBeta  = (const float*)d_in[4];
  const float* W       = (const float*)d_in[5];
  const float* bvec    = (const float*)d_in[6];
  float* Out = (float*)d_out;

  float* ws      = (float*)d_ws;
  float* frs     = ws;  ws += (size_t)B_ * NR;        // 262144
  float* muA     = ws;  ws += B_;                     // 2048
  float* istdA   = ws;  ws += B_;                     // 2048
  float* scaleA  = ws;  ws += IN_DIM * NR;            // 32768
  float* A2      = ws;  ws += OUT_DIM;                // 64
  float* Cc      = ws;  ws += OUT_DIM;                // 64
  float* Gfull   = ws;  ws += (size_t)OUT_DIM * DD;   // 2105344 (16B-aligned offset)
  float* partial = ws;                                // RG * B * OUT_DIM = 1048576

  // allow >64KB dynamic LDS for the double-buffered GEMM (HW supports 320KB/WG)
  (void)hipFuncSetAttribute((const void*)k_gemm_wmma,
                            hipFuncAttributeMaxDynamicSharedMemorySize, GEMM_LDS_BYTES);

  k_scale<<<(IN_DIM * NR) / 256, 256, 0, stream>>>(sigmas, scaleA);
  dim3 ggw((DD + 255) / 256, OUT_DIM);
  k_gw<<<ggw, 256, 0, stream>>>(lnGamma, W, Gfull);
  k_wsum<<<OUT_DIM, 256, 0, stream>>>(lnGamma, lnBeta, W, bvec, A2, Cc);
  k_firing<<<B_ / ROWS, 128, 0, stream>>>(X, centers, scaleA, frs, muA, istdA);
  dim3 g4(B_ / BT, RG);
  k_gemm_wmma<<<g4, 256, GEMM_LDS_BYTES, stream>>>(X, frs, Gfull, partial);
  k_finalize<<<(B_ * OUT_DIM) / 256, 256, 0, stream>>>(partial, frs, Gfull, muA, istdA, A2, Cc, Out);
}